// Project_17575006175537
// MI455X (gfx1250) — compile-verified
//
#include <hip/hip_runtime.h>

typedef float v2f __attribute__((ext_vector_type(2)));
typedef float v8f __attribute__((ext_vector_type(8)));

#define DMODEL 1024
#define DPROJ  256
#define NTOK   65536
#define TM     128          // tokens per block
#define KC     64           // K-chunk staged in LDS
#define KCP    68           // padded LDS row stride (272B: 16B aligned, bank stride 4)
#define LNEPS  1e-5f

// W-chunk staging via CDNA5 async global->LDS copy (ASYNCcnt tracked).
// Each thread issues 16 b128 transfers: rows p = (tid>>4)+16i, cols qq*4..qq*4+3.
//   global byte stride per i: 16 rows * 4096 B = 65536
//   LDS    byte stride per i: 16 rows * KCP*4 = 4352
__device__ __forceinline__ void stage_async(const float* __restrict__ wchunk,
                                            unsigned g_base, unsigned l_base) {
#pragma unroll
    for (int i = 0; i < 16; ++i) {
        unsigned go = g_base + (unsigned)(i * 65536);
        unsigned lo = l_base + (unsigned)(i * 4352);
        asm volatile("global_load_async_to_lds_b128 %0, %1, %2"
                     :: "v"(lo), "v"(go), "s"(wchunk) : "memory");
    }
}

__device__ __forceinline__ void wait_async0() {
    asm volatile("s_wait_asynccnt 0x0" ::: "memory");
}

__launch_bounds__(256)
__global__ void fused_ln_gemm_ln(const float* __restrict__ X,
                                 const float* __restrict__ g1,
                                 const float* __restrict__ b1,
                                 const float* __restrict__ W,
                                 const float* __restrict__ bias,
                                 const float* __restrict__ g2,
                                 const float* __restrict__ b2,
                                 float* __restrict__ Out)
{
    extern __shared__ __align__(16) float smem[];
    float* s_w0   = smem;                    // DPROJ*KCP
    float* s_w1   = s_w0 + DPROJ * KCP;      // DPROJ*KCP
    float* s_g1   = s_w1 + DPROJ * KCP;      // DMODEL
    float* s_b1   = s_g1 + DMODEL;           // DMODEL
    float* s_g2   = s_b1 + DMODEL;           // DPROJ
    float* s_b2   = s_g2 + DPROJ;            // DPROJ
    float* s_bias = s_b2 + DPROJ;            // DPROJ
    float* s_mu   = s_bias + DPROJ;          // TM
    float* s_rs   = s_mu + TM;               // TM

    const int tid  = threadIdx.x;
    const int lane = tid & 31;
    const int wave = tid >> 5;
    const int l    = lane & 15;              // column-in-tile / row-in-A-tile
    const int h    = lane >> 4;              // half-wave select (K pair / row+8)
    const int mBase = blockIdx.x * TM + wave * 16;

    // per-thread staging offsets (bytes)
    const int p0 = tid >> 4;                 // 0..15 (row group)
    const int qq = tid & 15;                 // float4 within row-chunk
    const unsigned g_base  = (unsigned)(p0 * 4096 + qq * 16);
    const unsigned l_base0 = (unsigned)(uintptr_t)(s_w0 + p0 * KCP + (qq << 2));
    const unsigned l_base1 = (unsigned)(uintptr_t)(s_w1 + p0 * KCP + (qq << 2));

    // ---- phase 0: LN/bias params -> LDS ----
    for (int i = tid; i < DMODEL; i += 256) { s_g1[i] = g1[i]; s_b1[i] = b1[i]; }
    if (tid < DPROJ) { s_g2[tid] = g2[tid]; s_b2[tid] = b2[tid]; s_bias[tid] = bias[tid]; }

    // kick off async staging of W chunk 0 (overlaps with the stats pass)
    stage_async(W, g_base, l_base0);

    // ---- phase 1: LN1 stats for this wave's 16 tokens (coalesced float4) ----
    for (int t = 0; t < 16; ++t) {
        const float* row = X + (size_t)(mBase + t) * DMODEL;
        float s = 0.f, q = 0.f;
        #pragma unroll
        for (int j = 0; j < 8; ++j) {
            float4 v = *(const float4*)(row + j * 128 + lane * 4);
            s += v.x + v.y + v.z + v.w;
            q += v.x * v.x + v.y * v.y + v.z * v.z + v.w * v.w;
        }
        #pragma unroll
        for (int m = 16; m >= 1; m >>= 1) {
            s += __shfl_xor(s, m, 32);
            q += __shfl_xor(q, m, 32);
        }
        if (lane == 0) {
            float mu  = s * (1.f / DMODEL);
            float var = q * (1.f / DMODEL) - mu * mu;
            s_mu[wave * 16 + t] = mu;
            s_rs[wave * 16 + t] = rsqrtf(var + LNEPS);
        }
    }
    wait_async0();
    __syncthreads();

    // per-lane row constants for the A fragment (lane l = token row mBase+l)
    const float  mu_l = s_mu[wave * 16 + l];
    const float  rs_l = s_rs[wave * 16 + l];
    const float* xrow = X + (size_t)(mBase + l) * DMODEL;

    v8f acc[16];
    #pragma unroll
    for (int n = 0; n < 16; ++n) acc[n] = (v8f){0.f,0.f,0.f,0.f,0.f,0.f,0.f,0.f};

    // ---- phase 2: GEMM, K=1024 in 16 chunks of 64, double-buffered async W ----
    const int NCH = DMODEL / KC;     // 16
    for (int c = 0; c < NCH; ++c) {
        const float* swc = (c & 1) ? s_w1 : s_w0;
        const bool more = (c + 1 < NCH);
        if (more) {
            // stage next chunk into the other buffer while we compute
            stage_async(W + (c + 1) * KC, g_base, (c & 1) ? l_base0 : l_base1);
        }

        #pragma unroll
        for (int kk = 0; kk < KC; kk += 8) {
            const int k2a = c * KC + kk + 2 * h;   // global K index, step a
            const int k2b = k2a + 4;               // step b
            // A fragments: on-the-fly LayerNorm of x
            v2f xa = *(const v2f*)(xrow + k2a);
            v2f xb = *(const v2f*)(xrow + k2b);
            v2f ga = *(const v2f*)(s_g1 + k2a);
            v2f ba = *(const v2f*)(s_b1 + k2a);
            v2f gb = *(const v2f*)(s_g1 + k2b);
            v2f bb = *(const v2f*)(s_b1 + k2b);
            v2f a0, a1;
            a0.x = (xa.x - mu_l) * rs_l * ga.x + ba.x;
            a0.y = (xa.y - mu_l) * rs_l * ga.y + ba.y;
            a1.x = (xb.x - mu_l) * rs_l * gb.x + bb.x;
            a1.y = (xb.y - mu_l) * rs_l * gb.y + bb.y;

            // load all 32 B fragments first -> up to 32 DS loads in flight
            const float* wb0 = swc + kk + 2 * h;
            v2f bf0[16], bf1[16];
            #pragma unroll
            for (int nt = 0; nt < 16; ++nt) {
                const float* p = wb0 + (nt * 16 + l) * KCP;
                bf0[nt] = *(const v2f*)(p);
                bf1[nt] = *(const v2f*)(p + 4);
            }
            #pragma unroll
            for (int nt = 0; nt < 16; ++nt)
                acc[nt] = __builtin_amdgcn_wmma_f32_16x16x4_f32(
                    false, a0, false, bf0[nt], (short)0, acc[nt], false, false);
            #pragma unroll
            for (int nt = 0; nt < 16; ++nt)
                acc[nt] = __builtin_amdgcn_wmma_f32_16x16x4_f32(
                    false, a1, false, bf1[nt], (short)0, acc[nt], false, false);
        }

        if (more) wait_async0();   // next buffer fully staged
        __syncthreads();
    }

    // ---- phase 3: bias add (before LN2 stats) ----
    #pragma unroll
    for (int nt = 0; nt < 16; ++nt) {
        float bv = s_bias[nt * 16 + l];
        #pragma unroll
        for (int g = 0; g < 8; ++g) acc[nt][g] += bv;
    }

    // ---- phase 4: LN2 entirely in registers.
    // D layout: VGPR g = row (g + 8h), lane l = column; reduce across 16 nt tiles
    // in-lane, then across the 16-lane group (xor masks < 16 stay in the half).
    float mu2[8], rs2[8];
    #pragma unroll
    for (int g = 0; g < 8; ++g) {
        float s = 0.f, q = 0.f;
        #pragma unroll
        for (int nt = 0; nt < 16; ++nt) { float v = acc[nt][g]; s += v; q += v * v; }
        #pragma unroll
        for (int m = 8; m >= 1; m >>= 1) {
            s += __shfl_xor(s, m, 32);
            q += __shfl_xor(q, m, 32);
        }
        float mu  = s * (1.f / DPROJ);
        float var = q * (1.f / DPROJ) - mu * mu;
        mu2[g] = mu;
        rs2[g] = rsqrtf(var + LNEPS);
    }

    // ---- phase 5: normalize + store (lanes 0..15 contiguous cols -> coalesced) ----
    #pragma unroll
    for (int nt = 0; nt < 16; ++nt) {
        const int col = nt * 16 + l;
        const float gv = s_g2[col], bv = s_b2[col];
        #pragma unroll
        for (int g = 0; g < 8; ++g) {
            const int row = mBase + g + 8 * h;
            Out[(size_t)row * DPROJ + col] = (acc[nt][g] - mu2[g]) * rs2[g] * gv + bv;
        }
    }
}

extern "C" void kernel_launch(void* const* d_in, const int* in_sizes, int n_in,
                              void* d_out, int out_size, void* d_ws, size_t ws_size,
                              hipStream_t stream) {
    const float* X    = (const float*)d_in[0];
    const float* g1   = (const float*)d_in[1];
    const float* b1   = (const float*)d_in[2];
    const float* W    = (const float*)d_in[3];
    const float* bias = (const float*)d_in[4];
    const float* g2   = (const float*)d_in[5];
    const float* b2   = (const float*)d_in[6];
    float* Out = (float*)d_out;

    const size_t smem = (size_t)(2 * DPROJ * KCP + 2 * DMODEL + 3 * DPROJ + 2 * TM)
                        * sizeof(float);   // ~148 KB; CDNA5 allows up to 320 KB/WG
    hipFuncSetAttribute((const void*)fused_ln_gemm_ln,
                        hipFuncAttributeMaxDynamicSharedMemorySize, (int)smem);

    fused_ln_gemm_ln<<<NTOK / TM, 256, smem, stream>>>(X, g1, b1, W, bias, g2, b2, Out);
}